// SSDReduceBoundingBoxes_46651934769652
// MI455X (gfx1250) — compile-verified
//
#include <hip/hip_runtime.h>
#include <math.h>

// ---------------------------------------------------------------------------
// SSD box reduce + NMS, fully fused into ONE workgroup on ONE WGP (MI455X).
//
// Rationale (MI455X): total data ~320KB, worst-case compute ~5 MFLOP -> the
// kernel is latency/sync bound, not BW (23.3 TB/s) or FLOP bound. Greedy NMS
// is a serial chain over kept boxes; the cheapest synchronization on CDNA5 is
// the wave32 workgroup split-barrier, so we keep ALL state in the 320KB LDS
// of a single WGP (CDNA5-only: CDNA4 had 64KB/CU and could not hold this).
// CDNA5 paths used: global_load_async_to_lds_b32 (ASYNCcnt), s_wait_asynccnt,
// s_barrier_signal/s_barrier_wait (via __syncthreads), ds_* LDS ops.
// ---------------------------------------------------------------------------

#define NPAD     8192      // power-of-two >= total boxes (1600 + 6400 = 8000)
#define NTHREADS 1024      // 32 wave32 waves, one workgroup
#define PROB_TH  0.9f
#define NEGF     (-1.0e9f)
#define IMG_W    640.0f
#define IMG_H    480.0f

__device__ __forceinline__ unsigned int f32_orderable(float f) {
    // monotone float -> uint mapping (ascending)
    unsigned int b = __float_as_uint(f);
    return (b & 0x80000000u) ? ~b : (b ^ 0x80000000u);
}

__global__ __launch_bounds__(NTHREADS)
void ssd_nms_fused_kernel(const float* __restrict__ outs0,
                          const float* __restrict__ outs1,
                          int P0, int P1,
                          float* __restrict__ out)
{
    // ---- LDS layout: ~270KB, fits CDNA5's 320KB/WGP ----
    __shared__ float              s_prob[NPAD];                 //  32 KB (phase A)
    __shared__ unsigned long long s_key [NPAD];                 //  64 KB (sort keys)
    __shared__ float              s_x1[NPAD], s_y1[NPAD];       //  64 KB (sorted xyxy)
    __shared__ float              s_x2[NPAD], s_y2[NPAD];       //  64 KB
    __shared__ float              s_sc[NPAD];                   //  32 KB (sorted scores)
    __shared__ unsigned char      s_keep[NPAD];                 //   8 KB
    __shared__ int                s_nvalid;

    const int tid = threadIdx.x;
    const int N0 = P0 * P0;
    const int N1 = P1 * P1;
    const int N  = N0 + N1;           // 8000 for the reference shapes

    if (tid == 0) s_nvalid = 0;

    // =====================================================================
    // Phase A0: async-DMA the probability channels (channel 0 of each level)
    // from global into LDS. Tracked on ASYNCcnt (CDNA5 async-to-LDS path).
    // GVS addressing: saddr = uniform base (SGPR pair), vaddr = per-lane
    // byte offset, vdst = per-lane LDS byte address.
    // =====================================================================
    {
        const unsigned long long base0 = (unsigned long long)(uintptr_t)outs0;
        const unsigned long long base1 = (unsigned long long)(uintptr_t)outs1;
        for (int q = tid; q < N0; q += NTHREADS) {
            unsigned int lds  = (unsigned int)(unsigned long long)(uintptr_t)&s_prob[q];
            unsigned int goff = (unsigned int)(q * 4);
            asm volatile("global_load_async_to_lds_b32 %0, %1, %2"
                         :: "v"(lds), "v"(goff), "s"(base0) : "memory");
        }
        for (int q = tid; q < N1; q += NTHREADS) {
            unsigned int lds  = (unsigned int)(unsigned long long)(uintptr_t)&s_prob[N0 + q];
            unsigned int goff = (unsigned int)(q * 4);
            asm volatile("global_load_async_to_lds_b32 %0, %1, %2"
                         :: "v"(lds), "v"(goff), "s"(base1) : "memory");
        }
        asm volatile("s_wait_asynccnt 0x0" ::: "memory");
    }
    __syncthreads();   // all waves drained their ASYNCcnt -> LDS probs visible

    // =====================================================================
    // Phase A1: build 64-bit sort keys.
    //   key = (~orderable(valid ? score : NEG) << 32) | original_index
    // Ascending u64 sort == descending score with stable index tie-break,
    // exactly matching jnp.argsort(-where(valid, score, NEG)).
    // =====================================================================
    {
        int cnt = 0;
        for (int idx = tid; idx < NPAD; idx += NTHREADS) {
            if (idx < N) {
                float p   = s_prob[idx];
                bool  v   = p > PROB_TH;
                cnt      += v ? 1 : 0;
                float kf  = v ? p : NEGF;
                unsigned int u = f32_orderable(kf);
                s_key[idx] = ((unsigned long long)(~u) << 32) | (unsigned int)idx;
            } else {
                s_key[idx] = ~0ull;   // padding sorts to the very end
            }
        }
        if (cnt) atomicAdd(&s_nvalid, cnt);
    }

    // =====================================================================
    // Phase B: bitonic sort of 8192 u64 keys in LDS.
    // 91 steps, 4 compare-exchanges per thread per step.
    // =====================================================================
    for (int k = 2; k <= NPAD; k <<= 1) {
        for (int j = k >> 1; j > 0; j >>= 1) {
            __syncthreads();
            for (int p = tid; p < NPAD / 2; p += NTHREADS) {
                int i = ((p & ~(j - 1)) << 1) | (p & (j - 1));
                int l = i;
                int r = i | j;
                bool asc = (i & k) == 0;
                unsigned long long a = s_key[l];
                unsigned long long b = s_key[r];
                if ((a > b) == asc) { s_key[l] = b; s_key[r] = a; }
            }
        }
    }
    __syncthreads();

    // =====================================================================
    // Phase C: gather boxes in sorted order, apply the conditional scaling
    // (scale only where prob > th, as in _scale_level), convert to rounded
    // xyxy, store SoA in LDS. Global rereads hit L2 (inputs are 160KB).
    // =====================================================================
    for (int r = tid; r < N; r += NTHREADS) {
        unsigned long long key = s_key[r];
        unsigned int ord = (unsigned int)key;      // original box index
        float sc = 0.f, X1 = 0.f, Y1 = 0.f, X2 = 0.f, Y2 = 0.f;
        unsigned char kp = 0;
        if (ord < (unsigned int)N) {
            const float* src; int P, PP, k;
            if ((int)ord < N0) { src = outs0; P = P0; PP = N0; k = (int)ord; }
            else               { src = outs1; P = P1; PP = N1; k = (int)ord - N0; }
            float prob = src[k];
            float v1 = src[PP + k];
            float v2 = src[2 * PP + k];
            float v3 = src[3 * PP + k];
            float v4 = src[4 * PP + k];
            int   ii = k / P;
            int   jj = k - ii * P;
            float xps = IMG_W / (float)P;
            float yps = IMG_H / (float)P;
            bool  valid = prob > PROB_TH;
            float c1 = valid ? (v1 * xps + (float)ii * xps) : v1;
            float c2 = valid ? (v2 * yps + (float)jj * yps) : v2;
            float c3 = valid ? (v3 * IMG_W) : v3;
            float c4 = valid ? (v4 * IMG_H) : v4;
            X1 = rintf(c1);            // jnp.round == round-half-even == rintf (RNE)
            Y1 = rintf(c2);
            X2 = rintf(c3 + c1);
            Y2 = rintf(c4 + c2);
            sc = prob;
            kp = valid ? 1 : 0;
        }
        s_sc[r] = sc; s_x1[r] = X1; s_y1[r] = Y1; s_x2[r] = X2; s_y2[r] = Y2;
        s_keep[r] = kp;
    }
    __syncthreads();

    // =====================================================================
    // Phase D: greedy NMS. Valid boxes occupy exactly ranks [0, nvalid)
    // (invalid keys carry NEG and sort behind every valid score).
    // Outer loop is the serial chain; inner IoU sweep is parallel over all
    // 1024 lanes. Barrier only when box i actually suppresses (uniform
    // branch: all threads read the same LDS flag after a barrier).
    // =====================================================================
    {
        const int nv = s_nvalid;
        for (int i = 0; i < nv; ++i) {
            if (s_keep[i]) {
                float xi1 = s_x1[i], yi1 = s_y1[i];
                float xi2 = s_x2[i], yi2 = s_y2[i];
                float ai  = (xi2 - xi1) * (yi2 - yi1);
                for (int r = i + 1 + tid; r < nv; r += NTHREADS) {
                    if (s_keep[r]) {
                        float iw = fminf(xi2, s_x2[r]) - fmaxf(xi1, s_x1[r]);
                        float ih = fminf(yi2, s_y2[r]) - fmaxf(yi1, s_y1[r]);
                        iw = fmaxf(iw, 0.f);
                        ih = fmaxf(ih, 0.f);
                        float inter = iw * ih;
                        float aj  = (s_x2[r] - s_x1[r]) * (s_y2[r] - s_y1[r]);
                        float uni = ai + aj - inter;
                        float iou = (uni > 0.f) ? (inter / uni) : 0.f;
                        if (iou > 0.5f) s_keep[r] = 0;
                    }
                }
                __syncthreads();
            }
        }
    }
    __syncthreads();

    // =====================================================================
    // Phase E: emit [score, x1, y1, x2-x1, y2-y1] per rank, zeros if culled.
    // =====================================================================
    for (int r = tid; r < N; r += NTHREADS) {
        float* o = out + 5 * r;
        if (s_keep[r]) {
            float x1 = s_x1[r], y1 = s_y1[r];
            o[0] = s_sc[r];
            o[1] = x1;
            o[2] = y1;
            o[3] = s_x2[r] - x1;
            o[4] = s_y2[r] - y1;
        } else {
            o[0] = 0.f; o[1] = 0.f; o[2] = 0.f; o[3] = 0.f; o[4] = 0.f;
        }
    }
}

extern "C" void kernel_launch(void* const* d_in, const int* in_sizes, int n_in,
                              void* d_out, int out_size, void* d_ws, size_t ws_size,
                              hipStream_t stream) {
    (void)n_in; (void)d_ws; (void)ws_size; (void)out_size;
    const float* outs0 = (const float*)d_in[0];
    const float* outs1 = (const float*)d_in[1];
    // Derive grid sizes from input element counts (5*P*P each); np0/np1
    // scalar inputs (d_in[2], d_in[3]) are redundant with the shapes.
    int n0 = in_sizes[0] / 5;
    int n1 = in_sizes[1] / 5;
    int P0 = 1; while (P0 * P0 < n0) ++P0;   // 40
    int P1 = 1; while (P1 * P1 < n1) ++P1;   // 80
    ssd_nms_fused_kernel<<<1, NTHREADS, 0, stream>>>(outs0, outs1, P0, P1,
                                                     (float*)d_out);
}